// CompositeGNN_68436008895103
// MI455X (gfx1250) — compile-verified
//
#include <hip/hip_runtime.h>

// ---------------------------------------------------------------------------
// GraphSAGE (3 layers + linear head) for gfx1250 / MI455X.
//   - scatter/mean aggregation: bandwidth-bound, float4 gathers + f32 atomics
//   - dense layers: V_WMMA_F32_16X16X4_F32, one wave per 16x32 output strip,
//     weights pre-transposed so B pairs are single aligned float2 loads.
// N = 50000 (divisible by 16 -> no row guards), E = 800000.
// ---------------------------------------------------------------------------

typedef __attribute__((ext_vector_type(2))) float v2f;
typedef __attribute__((ext_vector_type(8))) float v8f;

#define N_NODES 50000
#define E_EDGES 800000
#define IN_C 128
#define HID 256
#define OUT_C 15

__global__ void zero_f32(float* __restrict__ p, long long n) {
  long long i = (long long)blockIdx.x * blockDim.x + threadIdx.x;
  if (i < n) p[i] = 0.0f;
}

__global__ void degree_kernel(const long long* __restrict__ ei,
                              float* __restrict__ cnt) {
  int e = blockIdx.x * blockDim.x + threadIdx.x;
  if (e < E_EDGES) {
    long long dst = ei[E_EDGES + e];
    atomicAdd(&cnt[dst], 1.0f);
  }
}

__global__ void invcnt_kernel(float* __restrict__ cnt) {
  int i = blockIdx.x * blockDim.x + threadIdx.x;
  if (i < N_NODES) cnt[i] = 1.0f / fmaxf(cnt[i], 1.0f);
}

// Wt[n*K + k] = W[k*N + n], zero-padded in n up to Npad.
__global__ void transpose_pad(const float* __restrict__ W, float* __restrict__ Wt,
                              int K, int N, int Npad) {
  int idx = blockIdx.x * blockDim.x + threadIdx.x;
  if (idx >= Npad * K) return;
  int n = idx / K;
  int k = idx - n * K;
  Wt[idx] = (n < N) ? W[k * N + n] : 0.0f;
}

// One thread per (edge, 4-feature chunk): float4 gather of h[src], 4 float
// atomics into agg[dst]. d is 128 or 256 so chunks are a power of two.
__global__ void scatter_kernel(const float* __restrict__ h,
                               const long long* __restrict__ ei,
                               float* __restrict__ agg,
                               int chunks_log2, int d) {
  long long tid = (long long)blockIdx.x * blockDim.x + threadIdx.x;
  int e = (int)(tid >> chunks_log2);
  if (e >= E_EDGES) return;
  int c = (int)(tid & ((1 << chunks_log2) - 1)) << 2;
  long long s  = ei[e];
  long long dt = ei[E_EDGES + e];
  const float4 v = *reinterpret_cast<const float4*>(h + (size_t)s * d + c);
  float* p = agg + (size_t)dt * d + c;
  atomicAdd(p + 0, v.x);
  atomicAdd(p + 1, v.y);
  atomicAdd(p + 2, v.z);
  atomicAdd(p + 3, v.w);
}

// out[tile] = act( (invc*A1)@W1 + A2@W2 + bias ),  W given transposed (n-major).
// One wave computes a 16 x (16*NT) strip.  fp32 WMMA, K accumulated 4/instr.
// A-layout (16x4 f32): lanes 0-15 -> M=lane, K={k,k+1}; lanes 16-31 -> K={k+2,k+3}
// B-layout (4x16 f32): lanes 0-15 -> N=lane, same K split.
// C/D (16x16 f32): vgpr i: lanes 0-15 -> (M=i, N=lane); lanes 16-31 -> (M=i+8, N=lane-16)
template <int K1, int K2, int NCOLS, int NT, bool RELU, bool GUARD>
__global__ void sage_gemm(const float* __restrict__ A1, const float* __restrict__ invc,
                          const float* __restrict__ W1t,
                          const float* __restrict__ A2, const float* __restrict__ W2t,
                          const float* __restrict__ bias, float* __restrict__ out,
                          int tiles_m, int ldc, int n_cols_out) {
  constexpr int WPR = NCOLS / (16 * NT);   // waves per row of tiles
  int wave_id = blockIdx.x * (blockDim.x >> 5) + (threadIdx.x >> 5);
  if (wave_id >= tiles_m * WPR) return;    // wave-uniform exit (EXEC all-1 at WMMA)
  int lane = threadIdx.x & 31;
  int tm = wave_id / WPR;
  int col0 = (wave_id - tm * WPR) * (16 * NT);

  int half = lane >> 4;        // 0: K pair {0,1}; 1: K pair {2,3}
  int l15  = lane & 15;
  int row  = tm * 16 + l15;
  int kb   = half << 1;

  v8f acc[NT];
#pragma unroll
  for (int t = 0; t < NT; ++t) acc[t] = {};

  if constexpr (K1 > 0) {
    float s = invc[row];                           // fold 1/deg into A load
    const float* arow = A1 + row * K1 + kb;        // 32-bit index math
    const float* wcol[NT];
#pragma unroll
    for (int t = 0; t < NT; ++t) wcol[t] = W1t + (col0 + t * 16 + l15) * K1 + kb;
#pragma unroll 8
    for (int k = 0; k < K1; k += 4) {
      float2 av = *reinterpret_cast<const float2*>(arow + k);
      v2f a; a.x = av.x * s; a.y = av.y * s;
#pragma unroll
      for (int t = 0; t < NT; ++t) {
        float2 bv = *reinterpret_cast<const float2*>(wcol[t] + k);
        v2f b; b.x = bv.x; b.y = bv.y;
        acc[t] = __builtin_amdgcn_wmma_f32_16x16x4_f32(false, a, false, b,
                                                       (short)0, acc[t], false, false);
      }
    }
  }
  if constexpr (K2 > 0) {
    const float* arow = A2 + row * K2 + kb;
    const float* wcol[NT];
#pragma unroll
    for (int t = 0; t < NT; ++t) wcol[t] = W2t + (col0 + t * 16 + l15) * K2 + kb;
#pragma unroll 8
    for (int k = 0; k < K2; k += 4) {
      float2 av = *reinterpret_cast<const float2*>(arow + k);
      v2f a; a.x = av.x; a.y = av.y;
#pragma unroll
      for (int t = 0; t < NT; ++t) {
        float2 bv = *reinterpret_cast<const float2*>(wcol[t] + k);
        v2f b; b.x = bv.x; b.y = bv.y;
        acc[t] = __builtin_amdgcn_wmma_f32_16x16x4_f32(false, a, false, b,
                                                       (short)0, acc[t], false, false);
      }
    }
  }

  // Epilogue: bias (+relu) and store. Guard only exists for the 15-wide head.
  int r0 = tm * 16 + half * 8;
#pragma unroll
  for (int t = 0; t < NT; ++t) {
    int col = col0 + t * 16 + l15;
    if (GUARD && col >= n_cols_out) continue;
    float bv = bias[col];
#pragma unroll
    for (int i = 0; i < 8; ++i) {
      float v = acc[t][i] + bv;
      if (RELU) v = fmaxf(v, 0.0f);
      out[(r0 + i) * ldc + col] = v;
    }
  }
}

extern "C" void kernel_launch(void* const* d_in, const int* in_sizes, int n_in,
                              void* d_out, int out_size, void* d_ws, size_t ws_size,
                              hipStream_t stream) {
  const float*     x  = (const float*)d_in[0];
  const long long* ei = (const long long*)d_in[1];
  const float* Wl1 = (const float*)d_in[2];
  const float* bl1 = (const float*)d_in[3];
  const float* Wr1 = (const float*)d_in[4];
  const float* Wl2 = (const float*)d_in[5];
  const float* bl2 = (const float*)d_in[6];
  const float* Wr2 = (const float*)d_in[7];
  const float* Wl3 = (const float*)d_in[8];
  const float* bl3 = (const float*)d_in[9];
  const float* Wr3 = (const float*)d_in[10];
  const float* Wo  = (const float*)d_in[11];
  const float* bo  = (const float*)d_in[12];
  float* out = (float*)d_out;

  // Workspace: cnt[N] | agg[N*256] | bufA[N*256] | bufB[N*256] | Wt buffers
  float* cnt  = (float*)d_ws;
  float* agg  = cnt + N_NODES;
  float* bufA = agg + (size_t)N_NODES * HID;
  float* bufB = bufA + (size_t)N_NODES * HID;
  float* wt   = bufB + (size_t)N_NODES * HID;
  float* Wl1t = wt;                       // 256*128
  float* Wr1t = Wl1t + HID * IN_C;        // 256*128
  float* Wl2t = Wr1t + HID * IN_C;        // 256*256
  float* Wr2t = Wl2t + HID * HID;
  float* Wl3t = Wr2t + HID * HID;
  float* Wr3t = Wl3t + HID * HID;
  float* Wot  = Wr3t + HID * HID;         // 16*256 (padded)

  const int T = 256;
  auto blocks = [](long long n, int t) { return (unsigned)((n + t - 1) / t); };

  const int tiles_m = N_NODES / 16;       // 3125 (exact)
  auto gemm_blocks = [&](int waves_per_row) {
    int total_waves = tiles_m * waves_per_row;
    return (unsigned)((total_waves + (T / 32) - 1) / (T / 32));
  };

  // -- weight transposes (tiny; once per launch) -----------------------------
  transpose_pad<<<blocks(HID * IN_C, T), T, 0, stream>>>(Wl1, Wl1t, IN_C, HID, HID);
  transpose_pad<<<blocks(HID * IN_C, T), T, 0, stream>>>(Wr1, Wr1t, IN_C, HID, HID);
  transpose_pad<<<blocks(HID * HID, T), T, 0, stream>>>(Wl2, Wl2t, HID, HID, HID);
  transpose_pad<<<blocks(HID * HID, T), T, 0, stream>>>(Wr2, Wr2t, HID, HID, HID);
  transpose_pad<<<blocks(HID * HID, T), T, 0, stream>>>(Wl3, Wl3t, HID, HID, HID);
  transpose_pad<<<blocks(HID * HID, T), T, 0, stream>>>(Wr3, Wr3t, HID, HID, HID);
  transpose_pad<<<blocks(16 * HID, T), T, 0, stream>>>(Wo, Wot, HID, OUT_C, 16);

  // -- degrees ---------------------------------------------------------------
  zero_f32<<<blocks(N_NODES, T), T, 0, stream>>>(cnt, N_NODES);
  degree_kernel<<<blocks(E_EDGES, T), T, 0, stream>>>(ei, cnt);
  invcnt_kernel<<<blocks(N_NODES, T), T, 0, stream>>>(cnt);

  // -- layer 1 (agg d=128, out 256) ------------------------------------------
  zero_f32<<<blocks((long long)N_NODES * IN_C, T), T, 0, stream>>>(agg, (long long)N_NODES * IN_C);
  scatter_kernel<<<blocks((long long)E_EDGES * (IN_C / 4), T), T, 0, stream>>>(x, ei, agg, 5, IN_C);
  sage_gemm<IN_C, IN_C, HID, 2, true, false>
      <<<gemm_blocks(HID / 32), T, 0, stream>>>(agg, cnt, Wl1t, x, Wr1t,
                                                bl1, bufA, tiles_m, HID, HID);

  // -- layer 2 ---------------------------------------------------------------
  zero_f32<<<blocks((long long)N_NODES * HID, T), T, 0, stream>>>(agg, (long long)N_NODES * HID);
  scatter_kernel<<<blocks((long long)E_EDGES * (HID / 4), T), T, 0, stream>>>(bufA, ei, agg, 6, HID);
  sage_gemm<HID, HID, HID, 2, true, false>
      <<<gemm_blocks(HID / 32), T, 0, stream>>>(agg, cnt, Wl2t, bufA, Wr2t,
                                                bl2, bufB, tiles_m, HID, HID);

  // -- layer 3 ---------------------------------------------------------------
  zero_f32<<<blocks((long long)N_NODES * HID, T), T, 0, stream>>>(agg, (long long)N_NODES * HID);
  scatter_kernel<<<blocks((long long)E_EDGES * (HID / 4), T), T, 0, stream>>>(bufB, ei, agg, 6, HID);
  sage_gemm<HID, HID, HID, 2, true, false>
      <<<gemm_blocks(HID / 32), T, 0, stream>>>(agg, cnt, Wl3t, bufB, Wr3t,
                                                bl3, bufA, tiles_m, HID, HID);

  // -- output head (no aggregation, no relu, 15 cols, store-guarded) ---------
  sage_gemm<0, HID, 16, 1, false, true>
      <<<gemm_blocks(1), T, 0, stream>>>(nullptr, nullptr, nullptr, bufA, Wot,
                                         bo, out, tiles_m, OUT_C, OUT_C);
}